// UMGMQuantizer_48421461295643
// MI455X (gfx1250) — compile-verified
//
#include <hip/hip_runtime.h>
#include <hip/hip_bf16.h>
#include <cstring>

// ---------------- problem dimensions (fixed by the reference) ----------------
#define C_   256
#define M_   8
#define K_   2048
#define D_   32
#define N_   2
#define H_   64
#define W_   64
#define H1_  32
#define W1_  32
#define H2_  16
#define W2_  16
#define P0_  (N_*H1_*W1_)   // 2048 stage-0 latent positions
#define P1_  (N_*H2_*W2_)   // 512  stage-1 latent positions

typedef float v2f __attribute__((ext_vector_type(2)));
typedef float v8f __attribute__((ext_vector_type(8)));

// ---------------- generic WMMA GEMM with fused epilogues ----------------
// C(MxN) = A(MxK) * B(KxN); A element (r,c) = A[r*lda + c*as_], B element (r,c) = B[r*ldb + c*bs_]
// MODE: 0 store, 1 accumulate, 2 logit epilogue, 3 store minus S, 4 conv-transpose scatter
// AUNIT: as_ == 1 (A rows K-contiguous)   BUNIT: ldb == 1 (B cols K-contiguous)
struct GemmParams {
  const float* A; long lda; long as_; long abatch;
  const float* B; long ldb; long bs_; long bbatch;
  float* C; long ldc;
  const float* S;                                   // mode 3
  float* logit; const float* x2; const float* c2;   // mode 2
  int HW; long Kq;                                  // mode 2
  float* scat; long rowstride; long nstride;        // mode 4
  int Hin; int Win; int Wout; int ky; int kx;       // mode 4
  int Mdim; int Ndim; int Kdim;
};

template<int MODE>
__device__ __forceinline__ void gemm_epilogue(const GemmParams& p, int zb, int row0,
                                              int hi, int col, const v8f& acc) {
  for (int i = 0; i < 8; ++i) {
    const int row = row0 + i + hi * 8;
    const float v = acc[i];
    if (MODE == 0) {
      p.C[(long)row * p.ldc + col] = v;
    } else if (MODE == 1) {
      p.C[(long)row * p.ldc + col] += v;
    } else if (MODE == 2) {
      // row = spatial position, col = code index, zb = group m
      const int n  = row / p.HW;
      const int rr = row % p.HW;
      const float dist = p.x2[(long)zb * p.Mdim + row]
                       + p.c2[(long)zb * p.Kq + col] - 2.0f * v;
      p.logit[(((long)n * M_ + zb) * p.HW + rr) * p.Kq + col] = __logf(dist);
    } else if (MODE == 3) {
      p.C[(long)row * p.ldc + col] = v - p.S[(long)row * p.ldc + col];
    } else { // MODE == 4
      // row = out channel, col = latent position; scatter with stride-2 upsample
      const int hw = p.Hin * p.Win;
      const int n  = col / hw;
      const int r  = col % hw;
      const int y  = r / p.Win, x = r % p.Win;
      const long coff = (long)n * p.nstride
                      + (long)(2 * y + p.ky) * p.Wout + (2 * x + p.kx);
      p.scat[(long)row * p.rowstride + coff] = v;
    }
  }
}

template<int MODE, bool AUNIT, bool BUNIT>
__global__ __launch_bounds__(128) void wmma_gemm_kernel(GemmParams p) {
  const int lane  = threadIdx.x & 31;
  const int wave  = threadIdx.x >> 5;
  const int col0  = (blockIdx.x * 4 + wave) * 32;   // two 16-col tiles per wave
  const int row0  = blockIdx.y * 16;
  const int zb    = blockIdx.z;                     // batch (codebook group m) for mode 2
  const int mr    = lane & 15;
  const int hi    = lane >> 4;                      // 0: lanes 0-15, 1: lanes 16-31
  const int ka    = hi * 2;                         // A/B fragment K offset per half-wave

  const float* A = p.A + (long)zb * p.abatch;
  const float* B = p.B + (long)zb * p.bbatch;

  v8f acc0 = {0.f,0.f,0.f,0.f,0.f,0.f,0.f,0.f};
  v8f acc1 = {0.f,0.f,0.f,0.f,0.f,0.f,0.f,0.f};

#if defined(__gfx1250__)
  {
    const long arow  = (long)(row0 + mr) * p.lda;
    const long bcol0 = (long)(col0 + mr) * p.bs_;
    const long bcol1 = (long)(col0 + 16 + mr) * p.bs_;
    for (int k = 0; k < p.Kdim; k += 4) {
      v2f a, b0, b1;
      if (AUNIT) {
        a = *(const v2f*)(A + arow + (k + ka));                 // 8B aligned: lda,k,ka even
      } else {
        a.x = A[arow + (long)(k + ka)     * p.as_];
        a.y = A[arow + (long)(k + ka + 1) * p.as_];
      }
      if (BUNIT) {
        b0 = *(const v2f*)(B + bcol0 + (k + ka));               // 8B aligned: bs_,k,ka even
        b1 = *(const v2f*)(B + bcol1 + (k + ka));
      } else {
        b0.x = B[(long)(k + ka)     * p.ldb + bcol0];
        b0.y = B[(long)(k + ka + 1) * p.ldb + bcol0];
        b1.x = B[(long)(k + ka)     * p.ldb + bcol1];
        b1.y = B[(long)(k + ka + 1) * p.ldb + bcol1];
      }
      // two independent WMMAs per A fragment (no D->A/B hazard between them)
      acc0 = __builtin_amdgcn_wmma_f32_16x16x4_f32(false, a, false, b0,
                                                   (short)0, acc0, false, false);
      acc1 = __builtin_amdgcn_wmma_f32_16x16x4_f32(false, a, false, b1,
                                                   (short)0, acc1, false, false);
    }
  }
#else
  // host-pass stub (never executed on device)
  for (int k = 0; k < p.Kdim; ++k) { acc0[0] += A[k] * B[k]; acc1[0] += A[k] * B[k]; }
#endif

  // C/D layout: VGPR i holds row (row0 + i) for lanes 0-15, (row0 + i + 8) for lanes 16-31
  gemm_epilogue<MODE>(p, zb, row0, hi, col0 + mr,      acc0);
  gemm_epilogue<MODE>(p, zb, row0, hi, col0 + 16 + mr, acc1);
}

// ---------------- im2col for non-overlapping 2x2 stride-2 conv ----------------
// Bm[(i*4 + ky*2 + kx) * P + p] = in[n*sn + i*sc + (2y+ky)*sy + (2x+kx)]
__global__ __launch_bounds__(256) void im2col2x2_kernel(const float* in, float* Bm,
    long sn, long sc, long sy, int Hq, int Wq, long total) {
  long t = (long)blockIdx.x * blockDim.x + threadIdx.x;
  if (t >= total) return;
  const long P = (long)N_ * Hq * Wq;
  const long pcol = t % P;
  const long rowi = t / P;
  const int q = (int)(rowi & 3), i = (int)(rowi >> 2);
  const int ky = q >> 1, kx = q & 1;
  const int hw = Hq * Wq;
  const int n = (int)(pcol / hw), r = (int)(pcol % hw);
  const int y = r / Wq, x = r % Wq;
  Bm[t] = in[(long)n * sn + (long)i * sc + (long)(2 * y + ky) * sy + (2 * x + kx)];
}

// x2[m*P+p] = sum_d z[(m*D+d)*P + p]^2
__global__ __launch_bounds__(256) void sqsum_strided_kernel(const float* z, float* out, int P) {
  int t = blockIdx.x * blockDim.x + threadIdx.x;
  if (t >= M_ * P) return;
  const int m = t / P, pp = t % P;
  const float* b = z + (long)m * D_ * P + pp;
  float s = 0.f;
  for (int d = 0; d < D_; ++d) { float u = b[(long)d * P]; s += u * u; }
  out[t] = s;
}

// c2[t] = sum_d cb[t*D + d]^2  (t over M*K)
__global__ __launch_bounds__(256) void sqsum_contig_kernel(const float* cb, float* out, int R) {
  int t = blockIdx.x * blockDim.x + threadIdx.x;
  if (t >= R) return;
  const float* b = cb + (long)t * D_;
  float s = 0.f;
  for (int d = 0; d < D_; ++d) s += b[d] * b[d];
  out[t] = s;
}

__device__ __forceinline__ unsigned hash3(unsigned a, unsigned b, unsigned c) {
  unsigned h = a * 0x9E3779B1u ^ b * 0x85EBCA77u ^ c * 0xC2B2AE3Du;
  h ^= h >> 15; h *= 0x2C1B3C6Du;
  h ^= h >> 12; h *= 0x297A2D39u;
  h ^= h >> 15;
  return h;
}

// Gumbel-argmax categorical sampler: one block per (m,p) row of K logits
__global__ __launch_bounds__(256) void sample_rows_kernel(const float* logit, int* idxbuf,
    int* codeOut, int P, int HW, unsigned seed) {
  __shared__ float sv[256];
  __shared__ int   si[256];
  const int r = blockIdx.x;           // r = m*P + p
  const int m = r / P, pp = r % P;
  const int n = pp / HW, rr = pp % HW;
  const long base = (((long)n * M_ + m) * HW + rr) * (long)K_;
  float best = -3.4e38f; int bi = 0;
  for (int k = threadIdx.x; k < K_; k += 256) {
    unsigned h = hash3(seed, (unsigned)r, (unsigned)k);
    float u = ((float)(h >> 8) + 0.5f) * (1.0f / 16777216.0f);
    float g = logit[base + k] - __logf(-__logf(u));
    if (g > best) { best = g; bi = k; }
  }
  sv[threadIdx.x] = best; si[threadIdx.x] = bi;
  __syncthreads();
  for (int s = 128; s > 0; s >>= 1) {
    if (threadIdx.x < s) {
      if (sv[threadIdx.x + s] > sv[threadIdx.x] ||
          (sv[threadIdx.x + s] == sv[threadIdx.x] && si[threadIdx.x + s] < si[threadIdx.x])) {
        sv[threadIdx.x] = sv[threadIdx.x + s];
        si[threadIdx.x] = si[threadIdx.x + s];
      }
    }
    __syncthreads();
  }
  if (threadIdx.x == 0) {
    idxbuf[r] = si[0];
    codeOut[((long)n * M_ + m) * HW + rr] = si[0];
  }
}

// deq[c*P+p] = cb[(m*K + idx[m*P+p])*D + d],  c = m*D + d
__global__ __launch_bounds__(256) void dequant_kernel(const float* cb, const int* idxbuf,
    float* deq, int P) {
  long t = (long)blockIdx.x * blockDim.x + threadIdx.x;
  if (t >= (long)C_ * P) return;
  const int c = (int)(t / P), pp = (int)(t % P);
  const int m = c / D_, d = c % D_;
  const int id = idxbuf[m * P + pp];
  deq[t] = cb[((long)m * K_ + id) * D_ + d];
}

// ---------------- host orchestration ----------------
static inline GemmParams gp0() { GemmParams p; memset(&p, 0, sizeof(p)); return p; }

template<int MODE, bool AUNIT, bool BUNIT>
static inline void launch_gemm(const GemmParams& p, int batches, hipStream_t s) {
  dim3 grid((unsigned)(p.Ndim / 128), (unsigned)(p.Mdim / 16), (unsigned)batches);
  wmma_gemm_kernel<MODE, AUNIT, BUNIT><<<grid, 128, 0, s>>>(p);
}

// workspace offsets (in floats)
#define OFF_B0      0L
#define OFF_Z0      2097152L
#define OFF_QH0     2621440L
#define OFF_X2_0    3145728L
#define OFF_C2_0    3162112L
#define OFF_IDX0    3178496L   // int
#define OFF_DEQ0    3194880L
#define OFF_X1      3719168L
#define OFF_B1      4243456L
#define OFF_Z1      4767744L
#define OFF_QH1     4898816L
#define OFF_X2_1    5029888L
#define OFF_C2_1    5033984L
#define OFF_IDX1    5050368L   // int
#define OFF_DEQ1    5054464L
#define OFF_Q1      5185536L
#define OFF_FORMER  5316608L
#define OFF_XHAT    5840896L

// d_out offsets (in elements): out, code0, code1, logit0, logit1
#define DO_OUT     0L
#define DO_CODE0   2097152L
#define DO_CODE1   2113536L
#define DO_LOGIT0  2117632L
#define DO_LOGIT1  35672064L

extern "C" void kernel_launch(void* const* d_in, const int* in_sizes, int n_in,
                              void* d_out, int out_size, void* d_ws, size_t ws_size,
                              hipStream_t stream) {
  const float* x      = (const float*)d_in[0];
  const float* cb0    = (const float*)d_in[1];
  const float* cb1    = (const float*)d_in[2];
  const float* enc_w0 = (const float*)d_in[3];
  const float* enc_w1 = (const float*)d_in[4];
  const float* qh_w0  = (const float*)d_in[5];
  const float* qh_w1  = (const float*)d_in[6];
  const float* lh_w0  = (const float*)d_in[7];
  const float* dq_w0  = (const float*)d_in[8];
  const float* dq_w1  = (const float*)d_in[9];
  const float* side_w0= (const float*)d_in[10];
  const float* rst_w0 = (const float*)d_in[11];
  const float* rst_w1 = (const float*)d_in[12];

  float* ws   = (float*)d_ws;
  float* outp = (float*)d_out;
  int*   code0 = (int*)(outp + DO_CODE0);
  int*   code1 = (int*)(outp + DO_CODE1);
  float* logit0 = outp + DO_LOGIT0;
  float* logit1 = outp + DO_LOGIT1;

  // ---- stage 0 encode ----
  { // im2col: x (NCHW) -> B0 [1024 x P0]
    long total = 1024L * P0_;
    im2col2x2_kernel<<<(unsigned)((total + 255) / 256), 256, 0, stream>>>(
        x, ws + OFF_B0, (long)C_ * H_ * W_, (long)H_ * W_, (long)W_, H1_, W1_, total);
  }
  { // z0 = enc_w0 [256x1024] * B0 [1024xP0]
    GemmParams p = gp0();
    p.A = enc_w0; p.lda = 1024; p.as_ = 1;
    p.B = ws + OFF_B0; p.ldb = P0_; p.bs_ = 1;
    p.C = ws + OFF_Z0; p.ldc = P0_;
    p.Mdim = C_; p.Ndim = P0_; p.Kdim = 1024;
    launch_gemm<0, true, false>(p, 1, stream);
  }
  { // qh0 = qh_w0 * z0
    GemmParams p = gp0();
    p.A = qh_w0; p.lda = C_; p.as_ = 1;
    p.B = ws + OFF_Z0; p.ldb = P0_; p.bs_ = 1;
    p.C = ws + OFF_QH0; p.ldc = P0_;
    p.Mdim = C_; p.Ndim = P0_; p.Kdim = C_;
    launch_gemm<0, true, false>(p, 1, stream);
  }
  sqsum_strided_kernel<<<(M_ * P0_ + 255) / 256, 256, 0, stream>>>(ws + OFF_QH0, ws + OFF_X2_0, P0_);
  sqsum_contig_kernel<<<(M_ * K_ + 255) / 256, 256, 0, stream>>>(cb0, ws + OFF_C2_0, M_ * K_);
  { // logits0: per m, inter[p,k] = sum_d qh0[(m*D+d)*P0+p] * cb0[(m*K+k)*D+d]; fused log-dist
    GemmParams p = gp0();
    p.A = ws + OFF_QH0; p.lda = 1; p.as_ = P0_; p.abatch = (long)D_ * P0_;
    p.B = cb0;          p.ldb = 1; p.bs_ = D_;  p.bbatch = (long)K_ * D_;
    p.logit = logit0; p.x2 = ws + OFF_X2_0; p.c2 = ws + OFF_C2_0;
    p.HW = H1_ * W1_; p.Kq = K_;
    p.Mdim = P0_; p.Ndim = K_; p.Kdim = D_;
    launch_gemm<2, false, true>(p, M_, stream);
  }
  sample_rows_kernel<<<M_ * P0_, 256, 0, stream>>>(logit0, (int*)(ws + OFF_IDX0), code0,
                                                   P0_, H1_ * W1_, 0xC0DE0001u);
  dequant_kernel<<<(C_ * P0_ + 255) / 256, 256, 0, stream>>>(cb0, (int*)(ws + OFF_IDX0),
                                                             ws + OFF_DEQ0, P0_);
  { // x1 = lh_w0 * z0 - deq0
    GemmParams p = gp0();
    p.A = lh_w0; p.lda = C_; p.as_ = 1;
    p.B = ws + OFF_Z0; p.ldb = P0_; p.bs_ = 1;
    p.C = ws + OFF_X1; p.ldc = P0_; p.S = ws + OFF_DEQ0;
    p.Mdim = C_; p.Ndim = P0_; p.Kdim = C_;
    launch_gemm<3, true, false>(p, 1, stream);
  }

  // ---- stage 1 encode ----
  {
    long total = 1024L * P1_;
    im2col2x2_kernel<<<(unsigned)((total + 255) / 256), 256, 0, stream>>>(
        ws + OFF_X1, ws + OFF_B1, (long)H1_ * W1_, (long)P0_, (long)W1_, H2_, W2_, total);
  }
  {
    GemmParams p = gp0();
    p.A = enc_w1; p.lda = 1024; p.as_ = 1;
    p.B = ws + OFF_B1; p.ldb = P1_; p.bs_ = 1;
    p.C = ws + OFF_Z1; p.ldc = P1_;
    p.Mdim = C_; p.Ndim = P1_; p.Kdim = 1024;
    launch_gemm<0, true, false>(p, 1, stream);
  }
  {
    GemmParams p = gp0();
    p.A = qh_w1; p.lda = C_; p.as_ = 1;
    p.B = ws + OFF_Z1; p.ldb = P1_; p.bs_ = 1;
    p.C = ws + OFF_QH1; p.ldc = P1_;
    p.Mdim = C_; p.Ndim = P1_; p.Kdim = C_;
    launch_gemm<0, true, false>(p, 1, stream);
  }
  sqsum_strided_kernel<<<(M_ * P1_ + 255) / 256, 256, 0, stream>>>(ws + OFF_QH1, ws + OFF_X2_1, P1_);
  sqsum_contig_kernel<<<(M_ * K_ + 255) / 256, 256, 0, stream>>>(cb1, ws + OFF_C2_1, M_ * K_);
  {
    GemmParams p = gp0();
    p.A = ws + OFF_QH1; p.lda = 1; p.as_ = P1_; p.abatch = (long)D_ * P1_;
    p.B = cb1;          p.ldb = 1; p.bs_ = D_;  p.bbatch = (long)K_ * D_;
    p.logit = logit1; p.x2 = ws + OFF_X2_1; p.c2 = ws + OFF_C2_1;
    p.HW = H2_ * W2_; p.Kq = K_;
    p.Mdim = P1_; p.Ndim = K_; p.Kdim = D_;
    launch_gemm<2, false, true>(p, M_, stream);
  }
  sample_rows_kernel<<<M_ * P1_, 256, 0, stream>>>(logit1, (int*)(ws + OFF_IDX1), code1,
                                                   P1_, H2_ * W2_, 0xC0DE0002u);
  dequant_kernel<<<(C_ * P1_ + 255) / 256, 256, 0, stream>>>(cb1, (int*)(ws + OFF_IDX1),
                                                             ws + OFF_DEQ1, P1_);

  // ---- decode: stage 1 first ----
  { // q1 = dq_w1 * deq1
    GemmParams p = gp0();
    p.A = dq_w1; p.lda = C_; p.as_ = 1;
    p.B = ws + OFF_DEQ1; p.ldb = P1_; p.bs_ = 1;
    p.C = ws + OFF_Q1; p.ldc = P1_;
    p.Mdim = C_; p.Ndim = P1_; p.Kdim = C_;
    launch_gemm<0, true, false>(p, 1, stream);
  }
  // former = conv_transpose(q1, rst_w1): 4 scatter GEMMs (one per ky,kx)
  for (int q = 0; q < 4; ++q) {
    GemmParams p = gp0();
    p.A = rst_w1 + q; p.lda = (long)C_ * 4; p.as_ = 4;
    p.B = ws + OFF_Q1; p.ldb = P1_; p.bs_ = 1;
    p.scat = ws + OFF_FORMER; p.rowstride = P0_; p.nstride = (long)H1_ * W1_;
    p.Hin = H2_; p.Win = W2_; p.Wout = W1_; p.ky = q >> 1; p.kx = q & 1;
    p.Mdim = C_; p.Ndim = P1_; p.Kdim = C_;
    launch_gemm<4, false, false>(p, 1, stream);
  }
  { // xhat = dq_w0 * deq0
    GemmParams p = gp0();
    p.A = dq_w0; p.lda = C_; p.as_ = 1;
    p.B = ws + OFF_DEQ0; p.ldb = P0_; p.bs_ = 1;
    p.C = ws + OFF_XHAT; p.ldc = P0_;
    p.Mdim = C_; p.Ndim = P0_; p.Kdim = C_;
    launch_gemm<0, true, false>(p, 1, stream);
  }
  { // xhat += side_w0 * former
    GemmParams p = gp0();
    p.A = side_w0; p.lda = C_; p.as_ = 1;
    p.B = ws + OFF_FORMER; p.ldb = P0_; p.bs_ = 1;
    p.C = ws + OFF_XHAT; p.ldc = P0_;
    p.Mdim = C_; p.Ndim = P0_; p.Kdim = C_;
    launch_gemm<1, true, false>(p, 1, stream);
  }
  // out = conv_transpose(xhat, rst_w0) directly into d_out (NCHW)
  for (int q = 0; q < 4; ++q) {
    GemmParams p = gp0();
    p.A = rst_w0 + q; p.lda = (long)C_ * 4; p.as_ = 4;
    p.B = ws + OFF_XHAT; p.ldb = P0_; p.bs_ = 1;
    p.scat = outp + DO_OUT; p.rowstride = (long)H_ * W_; p.nstride = (long)C_ * H_ * W_;
    p.Hin = H1_; p.Win = W1_; p.Wout = W_; p.ky = q >> 1; p.kx = q & 1;
    p.Mdim = C_; p.Ndim = P0_; p.Kdim = C_;
    launch_gemm<4, false, false>(p, 1, stream);
  }
}